// TracerModel_62182536511720
// MI455X (gfx1250) — compile-verified
//
#include <hip/hip_runtime.h>

// ---------------------------------------------------------------------------
// Cart-pole RK4 residual kernel for MI455X (gfx1250, wave32).
//
// Purely elementwise over 2,097,152 lanes: no contraction -> WMMA is
// inapplicable. Bandwidth-bound: 14x8MB read + 4x8MB write ~= 151 MB/call
// -> ~6.5 us floor at 23.3 TB/s HBM (and the working set fits the 192 MB L2
// across graph replays). VALU is co-dominant, so this version also minimizes
// VALU count: shared sin/cos turn-normalization (one v_mul per v_sin+v_cos
// pair via __builtin_amdgcn_sinf/cosf), exact-grid launch (no EXEC guard),
// unsigned float4 indexing for SADDR-form b128 loads, and constant-offset
// channel stores sharing one address VGPR.
// ---------------------------------------------------------------------------

#define KB 4096
#define KW 512
#define KN (KB * KW)            // elements per input plane
#define KN4 (KN / 4)            // float4s per input plane (524288)

#define C_DT  0.019999999552965164f
#define C_G   9.8100004196167f
#define C_SF  0.05999999865889549f   // sigma_force
#define C_SOM 0.10000000149011612f   // sigma_omega
#define C_STH 0.009999999776482582f  // sigma_theta
#define C_SV  0.019999999552965164f  // sigma_v
#define C_SX  0.009999999776482582f  // sigma_x
#define C_INV2PI 0.15915494309189535f

struct InPtrs  { const float* p[14]; };
struct ParPtrs { const float* p[12]; };  // a1,a2,a4,a5,a6,a7,a8,a10,a11,a12,a13,a14

// One RK4 derivative evaluation. a1u = a1*u and a12u = a12*u are
// stage-invariant and hoisted by the caller.
__device__ __forceinline__ void deriv(
    float a1u, float a12u,
    float a2, float a4g, float a5, float inv_a6,
    float a10g, float a7, float a11, float inv_a13,
    float v, float th, float om,
    float& dx, float& dv, float& dth, float& dom)
{
    const float tn = th * C_INV2PI;          // one normalization for both
    const float s = __builtin_amdgcn_sinf(tn);   // v_sin_f32
    const float c = __builtin_amdgcn_cosf(tn);   // v_cos_f32
    dx  = v;
    dv  = (a1u - a2 * v + a4g * s * c - a5 * om * om * s) * inv_a6;
    dth = om;
    dom = (a12u * c + a10g * s - a7 * om - a11 * v) * inv_a13;
}

__global__ __launch_bounds__(256) void rk4_residual_kernel(
    InPtrs in, ParPtrs par, float* __restrict__ out)
{
    // Exact grid: 2048 blocks x 256 threads x float4 == KN. No guard needed.
    const unsigned t4 = blockIdx.x * 256u + threadIdx.x;   // float4 index

    // --- uniform scalar params (s_load path) ---
    const float a1  = par.p[0][0];
    const float a2  = par.p[1][0];
    const float a4  = par.p[2][0];
    const float a5  = par.p[3][0];
    const float a6  = par.p[4][0];
    const float a7  = par.p[5][0];
    const float a8  = par.p[6][0];
    const float a10 = par.p[7][0];
    const float a11 = par.p[8][0];
    const float a12 = par.p[9][0];
    const float a13 = par.p[10][0];
    const float a14 = par.p[11][0];
    const float a4g  = a4 * C_G;
    const float a10g = a10 * C_G;
    const float inv_a6  = 1.0f / a6;
    const float inv_a13 = 1.0f / a13;

    // --- 14 x global_load_b128 (SGPR base + shared u32 VGPR offset) ---
    const float4 xpos = reinterpret_cast<const float4*>(in.p[0])[t4];
    const float4 xvel = reinterpret_cast<const float4*>(in.p[1])[t4];
    const float4 xang = reinterpret_cast<const float4*>(in.p[2])[t4];
    const float4 xom  = reinterpret_cast<const float4*>(in.p[3])[t4];
    const float4 act  = reinterpret_cast<const float4*>(in.p[4])[t4];
    const float4 n1   = reinterpret_cast<const float4*>(in.p[5])[t4];
    const float4 n2   = reinterpret_cast<const float4*>(in.p[6])[t4];
    const float4 n3   = reinterpret_cast<const float4*>(in.p[7])[t4];
    const float4 n4   = reinterpret_cast<const float4*>(in.p[8])[t4];
    const float4 n5   = reinterpret_cast<const float4*>(in.p[9])[t4];
    const float4 ypos = reinterpret_cast<const float4*>(in.p[10])[t4];
    const float4 yvel = reinterpret_cast<const float4*>(in.p[11])[t4];
    const float4 yang = reinterpret_cast<const float4*>(in.p[12])[t4];
    const float4 yom  = reinterpret_cast<const float4*>(in.p[13])[t4];

    float4 o0, o1, o2, o3;
    float* const       o0p = &o0.x;
    float* const       o1p = &o1.x;
    float* const       o2p = &o2.x;
    float* const       o3p = &o3.x;
    const float* const xp  = &xpos.x;
    const float* const vp  = &xvel.x;
    const float* const tp  = &xang.x;
    const float* const op  = &xom.x;
    const float* const ap  = &act.x;
    const float* const n1p = &n1.x;
    const float* const n2p = &n2.x;
    const float* const n3p = &n3.x;
    const float* const n4p = &n4.x;
    const float* const n5p = &n5.x;
    const float* const yxp = &ypos.x;
    const float* const yvp = &yvel.x;
    const float* const ytp = &yang.x;
    const float* const yop = &yom.x;

#pragma unroll
    for (int j = 0; j < 4; ++j) {
        const float x  = xp[j];
        const float v  = vp[j];
        const float th = tp[j];
        const float om = op[j];
        const float u  = ap[j] + C_SF * n5p[j];
        const float a1u  = a1 * u;    // stage-invariant
        const float a12u = a12 * u;   // stage-invariant

        float k1x, k1v, k1t, k1o;
        float k2x, k2v, k2t, k2o;
        float k3x, k3v, k3t, k3o;
        float k4x, k4v, k4t, k4o;
        const float h2 = C_DT * 0.5f;

        deriv(a1u, a12u, a2, a4g, a5, inv_a6, a10g, a7, a11, inv_a13,
              v, th, om, k1x, k1v, k1t, k1o);
        deriv(a1u, a12u, a2, a4g, a5, inv_a6, a10g, a7, a11, inv_a13,
              v + h2 * k1v, th + h2 * k1t, om + h2 * k1o,
              k2x, k2v, k2t, k2o);
        deriv(a1u, a12u, a2, a4g, a5, inv_a6, a10g, a7, a11, inv_a13,
              v + h2 * k2v, th + h2 * k2t, om + h2 * k2o,
              k3x, k3v, k3t, k3o);
        deriv(a1u, a12u, a2, a4g, a5, inv_a6, a10g, a7, a11, inv_a13,
              v + C_DT * k3v, th + C_DT * k3t, om + C_DT * k3o,
              k4x, k4v, k4t, k4o);

        const float w6 = C_DT / 6.0f;
        const float x1  = x  + w6 * (k1x + 2.0f * k2x + 2.0f * k3x + k4x);
        const float v1  = v  + w6 * (k1v + 2.0f * k2v + 2.0f * k3v + k4v);
        const float th1 = th + w6 * (k1t + 2.0f * k2t + 2.0f * k3t + k4t);
        const float om1 = om + w6 * (k1o + 2.0f * k2o + 2.0f * k3o + k4o);

        const float px  = x1  + C_SX  * n1p[j];
        const float pv  = v1  + C_SV  * n2p[j];
        const float pth = th1 + C_STH * n3p[j];
        const float pom = om1 + C_SOM * n4p[j];

        o0p[j] = px - yxp[j];
        o1p[j] = a8  * (pv  - yvp[j]);
        o2p[j] = pth - ytp[j];
        o3p[j] = a14 * (pom - yop[j]);
    }

    // Output [B,4,W]: out[(b*4 + c)*W + w]. In float4 units:
    //   idx4 = b*512 + c*128 + w4,  b = t4>>7, w4 = t4&127.
    // The four channel stores share one address VGPR with immediate offsets
    // (0 / 2048 / 4096 / 6144 bytes fit the 24-bit IOFFSET).
    const unsigned b  = t4 >> 7;
    const unsigned w4 = t4 & 127u;
    float4* const obase = reinterpret_cast<float4*>(out) + (size_t)b * 512u + w4;
    obase[0]   = o0;   // global_store_b128, off:0
    obase[128] = o1;   // off:2048
    obase[256] = o2;   // off:4096
    obase[384] = o3;   // off:6144
}

extern "C" void kernel_launch(void* const* d_in, const int* in_sizes, int n_in,
                              void* d_out, int out_size, void* d_ws, size_t ws_size,
                              hipStream_t stream) {
    (void)in_sizes; (void)out_size; (void)d_ws; (void)ws_size;

    InPtrs in;
    for (int k = 0; k < 14; ++k) in.p[k] = (const float*)d_in[k];

    // Params dict in setup_inputs() insertion order:
    //   a1,a2,a4,a5,a6,a7,a8,a10,a11,a12,a13,a14
    ParPtrs par;
    if (n_in >= 26) {
        // Each param passed as its own 1-element input.
        for (int k = 0; k < 12; ++k) par.p[k] = (const float*)d_in[14 + k];
    } else {
        // Params packed as a single flat 12-element array.
        const float* P = (const float*)d_in[14];
        for (int k = 0; k < 12; ++k) par.p[k] = P + k;
    }

    const int threads = 256;
    const int blocks  = KN4 / threads;   // 2048, exact tiling
    rk4_residual_kernel<<<blocks, threads, 0, stream>>>(in, par, (float*)d_out);
}